// TopK_33028298506892
// MI455X (gfx1250) — compile-verified
//
#include <hip/hip_runtime.h>
#include <stdint.h>

// ---------------------------------------------------------------------------
// Top-k (k=64) by |x| per row, B=4096 rows, D=16384 cols, fp32.
// Strategy: TDM (tensor_load_to_lds) stages each 64KB row in LDS once;
// 4-pass MSB-first radix select over |x| bit patterns finds the exact
// 32-bit threshold key; one coalesced float4 pass writes the output.
// Global traffic: exactly 1x read + 1x write  (roofline ~22us @ 23.3 TB/s).
// ---------------------------------------------------------------------------

#define ROW_D   16384
#define TOPK    64
#define NT      256
#define VPT     (ROW_D / 4 / NT)   // float4 elements per thread = 16

typedef __attribute__((ext_vector_type(4))) unsigned int v4u;
typedef __attribute__((ext_vector_type(4))) int          v4i;
typedef __attribute__((ext_vector_type(8))) int          v8i;

__global__ __launch_bounds__(NT) void topk_abs_kernel(const float* __restrict__ x,
                                                      float* __restrict__ out) {
    __shared__ float    row[ROW_D];     // 64 KB staged row
    __shared__ unsigned hist[256];      // radix histogram
    __shared__ unsigned s_prefix;       // refined threshold prefix
    __shared__ unsigned s_rem;          // how many still needed at/below prefix
    __shared__ unsigned s_tie;          // tie-slot allocator

    const unsigned tid   = threadIdx.x;
    const size_t   rbase = (size_t)blockIdx.x * (size_t)ROW_D;

    // ---- Stage row: global -> LDS via Tensor Data Mover (wave 0 only) ----
#if __has_builtin(__builtin_amdgcn_tensor_load_to_lds)
    if (tid < 32u) {
        uint64_t ga      = (uint64_t)(uintptr_t)(x + rbase);
        unsigned lds_off = (unsigned)(uintptr_t)row;   // flat addr low 32 = LDS offset

        v4u g0 = (v4u)0u;
        g0.x = 1u;                                     // count=1 (valid user D#)
        g0.y = lds_off;                                // lds_addr        [63:32]
        g0.z = (unsigned)(ga & 0xFFFFFFFFu);           // global_addr lo  [95:64]
        g0.w = (unsigned)((ga >> 32) & 0x01FFFFFFu)    // global_addr hi  [120:96]
             | 0x80000000u;                            // type=2 ("image")[127:126]

        v8i g1 = (v8i)0;
        g1.s0 = 0x00020000;     // data_size=2 (4B); workgroup_mask=0
        g1.s1 = 0x40000000;     // tensor_dim0 = 16384   (bits 48..63)
        g1.s2 = 0x00010000;     // tensor_dim1 = 1       (bits 80..95)
        g1.s3 = 0x40000000;     // tile_dim0   = 16384   (bits 112..127)
        g1.s4 = 0x00000001;     // tile_dim1   = 1       (bits 128..143)
        g1.s5 = (int)ROW_D;     // tensor_dim0_stride    (bits 160..191)

        v4i gz = (v4i)0;
#if __clang_major__ >= 23
        v8i gz8 = (v8i)0;
        __builtin_amdgcn_tensor_load_to_lds(g0, g1, gz, gz, gz8, 0);
#else
        __builtin_amdgcn_tensor_load_to_lds(g0, g1, gz, gz, 0);
#endif
        __builtin_amdgcn_s_wait_tensorcnt(0);
    }
#else
    // Fallback: cooperative vector copy global -> LDS
    {
        const float4* src = (const float4*)(x + rbase);
        float4*       dst = (float4*)row;
        #pragma unroll
        for (int j = 0; j < VPT; ++j) dst[tid + NT * j] = src[tid + NT * j];
    }
#endif
    if (tid == 0) { s_prefix = 0u; s_rem = TOPK; s_tie = 0u; }
    __syncthreads();

    // ---- 4x MSB-first radix-select on key = bits(x) & 0x7FFFFFFF ----
    const uint4* rv = (const uint4*)row;
    #pragma unroll 1
    for (int pass = 0; pass < 4; ++pass) {
        const int shift = 24 - 8 * pass;
        hist[tid] = 0u;                       // 256 threads == 256 bins
        __syncthreads();

        const unsigned prefix = s_prefix;
        const unsigned hmask  = (pass == 0) ? 0u : (0xFFFFFFFFu << (shift + 8));

        #pragma unroll
        for (int j = 0; j < VPT; ++j) {
            uint4 kv = rv[tid + NT * j];
            unsigned k0 = kv.x & 0x7FFFFFFFu;
            unsigned k1 = kv.y & 0x7FFFFFFFu;
            unsigned k2 = kv.z & 0x7FFFFFFFu;
            unsigned k3 = kv.w & 0x7FFFFFFFu;
            if ((k0 & hmask) == prefix) atomicAdd(&hist[(k0 >> shift) & 0xFFu], 1u);
            if ((k1 & hmask) == prefix) atomicAdd(&hist[(k1 >> shift) & 0xFFu], 1u);
            if ((k2 & hmask) == prefix) atomicAdd(&hist[(k2 >> shift) & 0xFFu], 1u);
            if ((k3 & hmask) == prefix) atomicAdd(&hist[(k3 >> shift) & 0xFFu], 1u);
        }
        __syncthreads();

        if (tid == 0) {
            unsigned rem = s_rem, cum = 0u;
            int b = 255;
            for (; b > 0; --b) {
                unsigned c = hist[b];
                if (cum + c >= rem) break;
                cum += c;
            }
            s_prefix = prefix | ((unsigned)b << shift);
            s_rem    = rem - cum;
        }
        __syncthreads();
    }

    // ---- Output pass: keep key > T; allocate (key == T) ties up to quota ----
    const unsigned T     = s_prefix;
    const unsigned quota = s_rem;
    const float4*  rf    = (const float4*)row;
    float4*        ov    = (float4*)(out + rbase);

    auto sel = [&](float f) -> float {
        unsigned kk = __float_as_uint(f) & 0x7FFFFFFFu;
        if (kk > T) return f;
        if (kk == T) {
            unsigned slot = atomicAdd(&s_tie, 1u);
            if (slot < quota) return f;
        }
        return 0.0f;
    };

    #pragma unroll
    for (int j = 0; j < VPT; ++j) {
        float4 v = rf[tid + NT * j];
        float4 o;
        o.x = sel(v.x);
        o.y = sel(v.y);
        o.z = sel(v.z);
        o.w = sel(v.w);
        ov[tid + NT * j] = o;
    }
}

extern "C" void kernel_launch(void* const* d_in, const int* in_sizes, int n_in,
                              void* d_out, int out_size, void* d_ws, size_t ws_size,
                              hipStream_t stream) {
    const float* x   = (const float*)d_in[0];
    float*       out = (float*)d_out;
    const int    B   = in_sizes[0] / ROW_D;   // 4096 rows
    topk_abs_kernel<<<B, NT, 0, stream>>>(x, out);
}